// BBoxHead_36077725286524
// MI455X (gfx1250) — compile-verified
//
#include <hip/hip_runtime.h>
#include <hip/hip_bf16.h>
#include <math.h>

// MI455X / gfx1250, wave32. fp32 GEMMs via V_WMMA_F32_16X16X4_F32 with
// LDS-tiled double-buffered pipelining: global loads for tile kt+1 are in
// flight (registers) while WMMAs consume LDS tile kt.

typedef __attribute__((ext_vector_type(2))) float v2f;
typedef __attribute__((ext_vector_type(8))) float v8f;

#define N_ROIS  2000
#define K_CONV1 12544   // 7*7*256
#define HID     1024
#define NCLS    81
#define NDELTA  324     // 81*4
#define BN_EPS  1e-3f

#define BM 128
#define BN 64
#define BK 32
#define PA 36   // A LDS row stride (dwords): 144B rows -> b128-aligned, conflict-free b64 frags
#define PB 34   // B^T LDS row stride (dwords): 8B-aligned b64 frag reads

// ---------------------------------------------------------------------------
// Fold bias + inference BN into per-column scale/shift.
// ---------------------------------------------------------------------------
__global__ void prep_scale_shift(
    const float* __restrict__ b1, const float* __restrict__ g1,
    const float* __restrict__ be1, const float* __restrict__ m1,
    const float* __restrict__ v1,
    const float* __restrict__ b2, const float* __restrict__ g2,
    const float* __restrict__ be2, const float* __restrict__ m2,
    const float* __restrict__ v2,
    const float* __restrict__ lb, const float* __restrict__ db,
    float* __restrict__ scale1, float* __restrict__ shift1,
    float* __restrict__ scale2, float* __restrict__ shift2,
    float* __restrict__ ones,  float* __restrict__ shiftL,
    float* __restrict__ shiftD) {
  int i = blockIdx.x * blockDim.x + threadIdx.x;
  if (i < HID) {
    float s1 = g1[i] * rsqrtf(v1[i] + BN_EPS);
    scale1[i] = s1;
    shift1[i] = (b1[i] - m1[i]) * s1 + be1[i];
    float s2 = g2[i] * rsqrtf(v2[i] + BN_EPS);
    scale2[i] = s2;
    shift2[i] = (b2[i] - m2[i]) * s2 + be2[i];
  }
  if (i < 512)    ones[i]  = 1.0f;
  if (i < NCLS)   shiftL[i] = lb[i];
  if (i < NDELTA) shiftD[i] = db[i];
}

// ---------------------------------------------------------------------------
// C = epilogue(A[MxK] * W[KxN]) with fused scale/shift (+optional ReLU).
// Block = 128 threads = 4 waves; block tile 128x64x32, double-buffered LDS.
// Wave w computes rows [w*32, w*32+32) of the block tile, all 64 cols.
// Requires K % 32 == 0 (holds: 12544, 1024).
// ---------------------------------------------------------------------------
__global__ __launch_bounds__(128) void gemm_wmma_f32(
    const float* __restrict__ A, const float* __restrict__ W,
    const float* __restrict__ scale, const float* __restrict__ shift,
    float* __restrict__ C, int M, int K, int N, int doRelu) {
  __shared__ float As[2][BM * PA];
  __shared__ float Bs[2][BN * PB];

  const int tid  = threadIdx.x;
  const int lane = tid & 31;
  const int wave = tid >> 5;
  const int half = lane >> 4;   // 0: lanes 0-15, 1: lanes 16-31
  const int l16  = lane & 15;

  const int n0     = blockIdx.x * BN;
  const int mBlock = blockIdx.y * BM;

  // global->register staging coordinates
  const int arow = tid >> 3;        // 0..15 (row group, 16-row stride x8)
  const int acol = (tid & 7) * 4;   // 0..28, float4 within 32-wide K slab
  const int brow = tid >> 4;        // 0..7  (k row, 8-row stride x4)
  const int bcol = (tid & 15) * 4;  // 0..60, float4 within 64-wide N slab

  const bool nVec = ((N & 3) == 0); // b128-safe W rows (1024, 324); 81 -> scalar

  float4 aReg[8];
  float4 bReg[4];

  const int numT = K / BK;

  auto loadGlobal = [&](int kt) {
    const int kbase = kt * BK;
#pragma unroll
    for (int i = 0; i < 8; ++i) {
      int r = mBlock + arow + 16 * i;
      if (r > M - 1) r = M - 1;                       // clamp; store is masked
      aReg[i] = *(const float4*)(A + (size_t)r * K + kbase + acol);
    }
#pragma unroll
    for (int i = 0; i < 4; ++i) {
      const int kr = kbase + brow + 8 * i;
      const int gc = n0 + bcol;
      if (nVec && gc + 3 < N) {
        bReg[i] = *(const float4*)(W + (size_t)kr * N + gc);
      } else {
        float4 v;
        v.x = (gc + 0 < N) ? W[(size_t)kr * N + gc + 0] : 0.0f;
        v.y = (gc + 1 < N) ? W[(size_t)kr * N + gc + 1] : 0.0f;
        v.z = (gc + 2 < N) ? W[(size_t)kr * N + gc + 2] : 0.0f;
        v.w = (gc + 3 < N) ? W[(size_t)kr * N + gc + 3] : 0.0f;
        bReg[i] = v;
      }
    }
  };

  auto storeLds = [&](int buf) {
#pragma unroll
    for (int i = 0; i < 8; ++i)
      *(float4*)&As[buf][(arow + 16 * i) * PA + acol] = aReg[i];
#pragma unroll
    for (int i = 0; i < 4; ++i) {                      // transpose B into Bs[col][k]
      const int kr = brow + 8 * i;
      Bs[buf][(bcol + 0) * PB + kr] = bReg[i].x;
      Bs[buf][(bcol + 1) * PB + kr] = bReg[i].y;
      Bs[buf][(bcol + 2) * PB + kr] = bReg[i].z;
      Bs[buf][(bcol + 3) * PB + kr] = bReg[i].w;
    }
  };

  v8f acc[2][4] = {};   // [m-subtile 16][n-tile 16] per wave: 32x64

  // Fragment layouts per CDNA5 ISA (CDNA5 ISA, 16x16x4 f32):
  //   A: lanes 0-15 hold (M=lane, K..K+1) in v0..v1; lanes 16-31 (K+2..K+3)
  //   B: lanes 0-15 hold (N=lane, rows K..K+1); lanes 16-31 rows (K+2..K+3)
  //   C: VGPR r, lanes 0-15: (M=r, N=lane); lanes 16-31: (M=r+8, N=lane-16)
  const int aBase0 = (wave * 32 + l16) * PA + 2 * half;
  const int aBase1 = aBase0 + 16 * PA;
  const int bBase  = l16 * PB + 2 * half;

  auto compute = [&](int buf) {
#pragma unroll
    for (int kk = 0; kk < BK; kk += 4) {
      v2f a0 = *(const v2f*)&As[buf][aBase0 + kk];
      v2f a1 = *(const v2f*)&As[buf][aBase1 + kk];
#pragma unroll
      for (int t = 0; t < 4; ++t) {
        v2f b = *(const v2f*)&Bs[buf][bBase + t * 16 * PB + kk];
        acc[0][t] = __builtin_amdgcn_wmma_f32_16x16x4_f32(
            false, a0, false, b, (short)0, acc[0][t], false, false);
        acc[1][t] = __builtin_amdgcn_wmma_f32_16x16x4_f32(
            false, a1, false, b, (short)0, acc[1][t], false, false);
      }
    }
  };

  loadGlobal(0);
  storeLds(0);
  __syncthreads();

  for (int kt = 0; kt < numT; ++kt) {
    const int cur = kt & 1;
    const bool more = (kt + 1 < numT);
    if (more) loadGlobal(kt + 1);   // global loads stay in flight during compute
    compute(cur);
    if (more) {
      __syncthreads();              // everyone done reading buffer cur^1
      storeLds(cur ^ 1);            // loadcnt wait lands here, after 64 WMMAs
      __syncthreads();
    }
  }

  // Epilogue: out = relu_opt(acc*scale[col] + shift[col]), masked stores.
#pragma unroll
  for (int ms = 0; ms < 2; ++ms) {
#pragma unroll
    for (int t = 0; t < 4; ++t) {
      const int col = n0 + t * 16 + l16;
      if (col < N) {
        const float sc = scale[col];
        const float sh = shift[col];
#pragma unroll
        for (int r = 0; r < 8; ++r) {
          const int row = mBlock + wave * 32 + ms * 16 + r + (half << 3);
          if (row < M) {
            float v = acc[ms][t][r] * sc + sh;
            if (doRelu) v = fmaxf(v, 0.0f);
            C[(size_t)row * N + col] = v;
          }
        }
      }
    }
  }
}

// ---------------------------------------------------------------------------
// Row softmax, one wave32 per row (N=81 -> 3 elems/lane max).
// ---------------------------------------------------------------------------
__global__ void softmax_rows(const float* __restrict__ logits,
                             float* __restrict__ probs, int M, int N) {
  int row  = blockIdx.x * (blockDim.x >> 5) + (threadIdx.x >> 5);
  int lane = threadIdx.x & 31;
  if (row >= M) return;
  const float* in = logits + (size_t)row * N;

  float mx = -3.4e38f;
  for (int j = lane; j < N; j += 32) mx = fmaxf(mx, in[j]);
#pragma unroll
  for (int off = 16; off > 0; off >>= 1) mx = fmaxf(mx, __shfl_xor(mx, off, 32));

  float sum = 0.0f;
  for (int j = lane; j < N; j += 32) sum += __expf(in[j] - mx);
#pragma unroll
  for (int off = 16; off > 0; off >>= 1) sum += __shfl_xor(sum, off, 32);

  float inv = 1.0f / sum;
  for (int j = lane; j < N; j += 32)
    probs[(size_t)row * N + j] = __expf(in[j] - mx) * inv;
}

// ---------------------------------------------------------------------------
extern "C" void kernel_launch(void* const* d_in, const int* in_sizes, int n_in,
                              void* d_out, int out_size, void* d_ws,
                              size_t ws_size, hipStream_t stream) {
  const float* pooled = (const float*)d_in[0];
  const float* w1  = (const float*)d_in[1];
  const float* b1  = (const float*)d_in[2];
  const float* g1  = (const float*)d_in[3];
  const float* be1 = (const float*)d_in[4];
  const float* mn1 = (const float*)d_in[5];
  const float* vr1 = (const float*)d_in[6];
  const float* w2  = (const float*)d_in[7];
  const float* b2  = (const float*)d_in[8];
  const float* g2  = (const float*)d_in[9];
  const float* be2 = (const float*)d_in[10];
  const float* mn2 = (const float*)d_in[11];
  const float* vr2 = (const float*)d_in[12];
  const float* wl  = (const float*)d_in[13];
  const float* bl  = (const float*)d_in[14];
  const float* wd  = (const float*)d_in[15];
  const float* bd  = (const float*)d_in[16];

  float* out = (float*)d_out;   // [logits 162000 | probs 162000 | deltas 648000]
  float* ws  = (float*)d_ws;    // needs ~16.5 MB

  float* H1     = ws;                   // 2000*1024
  float* H2     = ws + 2048000;         // 2000*1024
  float* scale1 = ws + 4096000;         // 1024
  float* shift1 = scale1 + 1024;
  float* scale2 = shift1 + 1024;
  float* shift2 = scale2 + 1024;
  float* ones   = shift2 + 1024;        // 512 (identity scale for heads)
  float* shiftL = ones + 512;           // 81 (slot 128)
  float* shiftD = shiftL + 128;         // 324

  prep_scale_shift<<<4, 256, 0, stream>>>(b1, g1, be1, mn1, vr1,
                                          b2, g2, be2, mn2, vr2,
                                          bl, bd,
                                          scale1, shift1, scale2, shift2,
                                          ones, shiftL, shiftD);

  const int gy = (N_ROIS + BM - 1) / BM;   // 16

  // GEMM1: [2000 x 12544] x [12544 x 1024] -> H1, fused BN1+ReLU
  gemm_wmma_f32<<<dim3(HID / BN, gy), 128, 0, stream>>>(
      pooled, w1, scale1, shift1, H1, N_ROIS, K_CONV1, HID, 1);

  // GEMM2: [2000 x 1024] x [1024 x 1024] -> H2, fused BN2+ReLU
  gemm_wmma_f32<<<dim3(HID / BN, gy), 128, 0, stream>>>(
      H1, w2, scale2, shift2, H2, N_ROIS, HID, HID, 1);

  // Logits head: [2000 x 1024] x [1024 x 81] -> out[0:162000], bias only
  gemm_wmma_f32<<<dim3((NCLS + BN - 1) / BN, gy), 128, 0, stream>>>(
      H2, wl, ones, shiftL, out, N_ROIS, HID, NCLS, 0);

  // Delta head: [2000 x 1024] x [1024 x 324] -> out[324000:972000], bias only
  gemm_wmma_f32<<<dim3((NDELTA + BN - 1) / BN, gy), 128, 0, stream>>>(
      H2, wd, ones, shiftD, out + 2 * N_ROIS * NCLS, N_ROIS, HID, NDELTA, 0);

  // Softmax over logits -> out[162000:324000]
  softmax_rows<<<(N_ROIS + 7) / 8, 256, 0, stream>>>(out, out + N_ROIS * NCLS,
                                                     N_ROIS, NCLS);
}